// NeuralGPKernel_12163347383245
// MI455X (gfx1250) — compile-verified
//
#include <hip/hip_runtime.h>
#include <math.h>

typedef __attribute__((ext_vector_type(2))) float v2f;
typedef __attribute__((ext_vector_type(8))) float v8f;

#define B_      2
#define N_O_    512
#define N_Q_    512
#define LATENT_ 256
#define HEADS_  8
#define HDIM_   32
#define HD_     256
#define ODIM_   128

static __device__ __forceinline__ v8f wmma_f32(v2f a, v2f b, v8f c) {
  // V_WMMA_F32_16X16X4_F32 : D = A(16x4) * B(4x16) + C(16x16)
  return __builtin_amdgcn_wmma_f32_16x16x4_f32(false, a, false, b, (short)0, c,
                                               false, false);
}

static __device__ __forceinline__ v8f vzero8() {
  v8f z = {0.f, 0.f, 0.f, 0.f, 0.f, 0.f, 0.f, 0.f};
  return z;
}

// ---------------------------------------------------------------------------
// K0: pad kw1 [9,256] -> [12,256], kw2 [256,8] -> [256,16]
// ---------------------------------------------------------------------------
__global__ __launch_bounds__(256) void k_init(const float* __restrict__ kw1,
                                              const float* __restrict__ kw2,
                                              float* __restrict__ kw1p,
                                              float* __restrict__ kw2p) {
  int t = threadIdx.x;
  for (int f = t; f < 12 * 256; f += 256) {
    int k = f >> 8;
    kw1p[f] = (k < 9) ? kw1[f] : 0.f;  // same flat layout for k < 9
  }
  for (int f = t; f < 256 * 16; f += 256) {
    int k = f >> 4, h = f & 15;
    kw2p[f] = (h < HEADS_) ? kw2[k * HEADS_ + h] : 0.f;
  }
}

// ---------------------------------------------------------------------------
// K1: v = relu(h_obs @ fw1 + fb1) @ fw2 + fb2        rows: [B*N_O, 256]
// block: 256 thr (8 waves), 16 rows per block; wave w -> N-tiles {2w,2w+1}
// ---------------------------------------------------------------------------
__global__ __launch_bounds__(256) void k_feature_mlp(
    const float* __restrict__ h_obs, const float* __restrict__ fw1,
    const float* __restrict__ fb1, const float* __restrict__ fw2,
    const float* __restrict__ fb2, float* __restrict__ v_out) {
  __shared__ float X[16 * 256];
  __shared__ float H[16 * 256];
  const int t = threadIdx.x;
  const long row0 = (long)blockIdx.x * 16;

#pragma unroll
  for (int i = 0; i < 16; ++i) {
    int f = i * 256 + t;
    X[f] = h_obs[row0 * 256 + f];
  }
  __syncthreads();

  const int w = t >> 5, l = t & 31, lm = l & 15, lg = l >> 4;

  // layer 1
  for (int nt = 2 * w; nt < 2 * w + 2; ++nt) {
    v8f c = vzero8();
    const int n = nt * 16 + lm;
    for (int kb = 0; kb < 64; ++kb) {
      int k0 = kb * 4 + 2 * lg;
      v2f a = {X[lm * 256 + k0], X[lm * 256 + k0 + 1]};
      v2f b = {fw1[k0 * 256 + n], fw1[(k0 + 1) * 256 + n]};
      c = wmma_f32(a, b, c);
    }
    float bias = fb1[n];
#pragma unroll
    for (int j = 0; j < 8; ++j) {
      int m = j + lg * 8;
      float h = c[j] + bias;
      H[m * 256 + n] = h > 0.f ? h : 0.f;
    }
  }
  __syncthreads();

  // layer 2
  for (int nt = 2 * w; nt < 2 * w + 2; ++nt) {
    v8f c = vzero8();
    const int n = nt * 16 + lm;
    for (int kb = 0; kb < 64; ++kb) {
      int k0 = kb * 4 + 2 * lg;
      v2f a = {H[lm * 256 + k0], H[lm * 256 + k0 + 1]};
      v2f b = {fw2[k0 * 256 + n], fw2[(k0 + 1) * 256 + n]};
      c = wmma_f32(a, b, c);
    }
    float bias = fb2[n];
#pragma unroll
    for (int j = 0; j < 8; ++j) {
      int m = j + lg * 8;
      v_out[(row0 + m) * 256 + n] = c[j] + bias;
    }
  }
}

// ---------------------------------------------------------------------------
// K2: fused pair MLP -> logits[b][h][q][o]
// grid: B x 32 q-tiles x 8 o-groups (512 blocks); block: 8 waves.
// wave w owns queries {q0+2w, q0+2w+1}; each block does 4 o-chunks of 16.
// ki features synthesized directly in WMMA A layout; layer-1 B fragments and
// bias hoisted into registers (invariant over q/o); hidden tile transposed
// C->A layout through a per-wave 16x17 LDS buffer.
// ---------------------------------------------------------------------------
__global__ __launch_bounds__(256) void k_pair_mlp(
    const float* __restrict__ pos_obs, const float* __restrict__ pos_query,
    const float* __restrict__ log_sigma, const float* __restrict__ kb1,
    const float* __restrict__ kb2, const float* __restrict__ kw1p,
    const float* __restrict__ kw2p, float* __restrict__ logits) {
  __shared__ float W1[12 * 256];
  __shared__ float W2[256 * 16];
  __shared__ float T[8 * 16 * 17];  // per-wave transpose buffer
  __shared__ float D2[8 * 16];      // per-wave dist^2 staging
  __shared__ float SIG[8], KB2S[8];

  const int t = threadIdx.x;
  for (int f = t; f < 12 * 256; f += 256) W1[f] = kw1p[f];
  for (int f = t; f < 256 * 16; f += 256) W2[f] = kw2p[f];
  if (t < HEADS_) {
    float s = expf(log_sigma[t]);
    SIG[t] = s * s + 1e-6f;
    KB2S[t] = kb2[t];
  }
  __syncthreads();

  const int og = blockIdx.x & 7;          // o-group: 4 chunks of 16 obs
  const int qt = (blockIdx.x >> 3) & 31;  // q-tile
  const int b = blockIdx.x >> 8;
  const int q0 = qt * 16;
  const int w = t >> 5, l = t & 31, lm = l & 15, lg = l >> 4;
  float* Tw = T + w * 16 * 17;

  // hoist layer-1 B fragments + bias into registers (48 v2f + 16 f32)
  v2f w1r[48];
  float b1r[16];
#pragma unroll
  for (int nt = 0; nt < 16; ++nt) {
    const int n = nt * 16 + lm;
    b1r[nt] = kb1[n];
#pragma unroll
    for (int kb = 0; kb < 3; ++kb) {
      int k0 = kb * 4 + 2 * lg;
      w1r[nt * 3 + kb] = (v2f){W1[k0 * 256 + n], W1[(k0 + 1) * 256 + n]};
    }
  }

  for (int qi = 0; qi < 2; ++qi) {
    const int q = q0 + 2 * w + qi;
    const float pq0 = pos_query[((long)b * N_Q_ + q) * 3 + 0];
    const float pq1 = pos_query[((long)b * N_Q_ + q) * 3 + 1];
    const float pq2 = pos_query[((long)b * N_Q_ + q) * 3 + 2];

    for (int ci = 0; ci < 4; ++ci) {
      const int ch = og * 4 + ci;
      const int o = ch * 16 + lm;  // lane's o for the A operand (pair row)
      const float po0 = pos_obs[((long)b * N_O_ + o) * 3 + 0];
      const float po1 = pos_obs[((long)b * N_O_ + o) * 3 + 1];
      const float po2 = pos_obs[((long)b * N_O_ + o) * 3 + 2];
      const float r0 = pq0 - po0, r1 = pq1 - po1, r2 = pq2 - po2;
      const float d2 = r0 * r0 + r1 * r1 + r2 * r2;
      if (l < 16) D2[w * 16 + lm] = d2;

      // ki features in A layout, K pairs per half-wave:
      // kb=0: K {0,1}|{2,3}  kb=1: K {4,5}|{6,7}  kb=2: K {8,9}|{10,11}
      v2f aF0 = lg ? (v2f){pq2, po0} : (v2f){pq0, pq1};
      v2f aF1 = lg ? (v2f){r0, r1} : (v2f){po1, po2};
      v2f aF2 = lg ? (v2f){0.f, 0.f} : (v2f){r2, 0.f};

      v8f cd = vzero8();  // layer-2 accumulator (delta, heads padded to 16)

#pragma unroll
      for (int nt = 0; nt < 16; ++nt) {
        // ---- layer 1: one 16x16 hidden tile (B operands from registers)
        v8f c = vzero8();
        c = wmma_f32(aF0, w1r[nt * 3 + 0], c);
        c = wmma_f32(aF1, w1r[nt * 3 + 1], c);
        c = wmma_f32(aF2, w1r[nt * 3 + 2], c);
        float bias = b1r[nt];
#pragma unroll
        for (int j = 0; j < 8; ++j) {  // C layout -> LDS (pair, hiddencol)
          int m = j + lg * 8;
          float h = c[j] + bias;
          Tw[m * 17 + lm] = h > 0.f ? h : 0.f;
        }
        // ---- layer 2 partial: K-slice [nt*16, nt*16+16)
#pragma unroll
        for (int kk = 0; kk < 4; ++kk) {
          int k0 = kk * 4 + 2 * lg;
          v2f a = {Tw[lm * 17 + k0], Tw[lm * 17 + k0 + 1]};
          int kg = nt * 16 + k0;
          v2f bb = {W2[kg * 16 + lm], W2[(kg + 1) * 16 + lm]};
          cd = wmma_f32(a, bb, cd);
        }
      }

      // logits: lane head = lm, pair(o) = j + 8*lg
      if (lm < HEADS_) {
        float s2 = SIG[lm];
        float kbias = KB2S[lm];
#pragma unroll
        for (int j = 0; j < 8; ++j) {
          int ol = j + lg * 8;
          float d2v = D2[w * 16 + ol];
          float lrbf = logf(expf(-d2v / s2) + 1e-8f);
          long idx = ((((long)b * HEADS_ + lm) * N_Q_) + q) * N_O_ + ch * 16 + ol;
          logits[idx] = lrbf + cd[j] + kbias;
        }
      }
    }
  }
}

// ---------------------------------------------------------------------------
// K3: softmax over o + E[v], E[v^2] via P@V and P@(V*V)
// block: (b, h, 32 queries). LDS holds P[32,512] (dynamic, 64 KB).
// 8 waves: (which={mean,sq}) x (mtile 0/1) x (ntile 0/1)
// ---------------------------------------------------------------------------
__global__ __launch_bounds__(256) void k_attn(const float* __restrict__ logits,
                                              const float* __restrict__ v,
                                              float* __restrict__ hq,
                                              float* __restrict__ ev2) {
  extern __shared__ float L[];  // 32 * 512
  const int t = threadIdx.x;
  const int bh = blockIdx.x >> 4;  // 16 q-tiles of 32
  const int b = bh >> 3, h = bh & 7;
  const int q0 = (blockIdx.x & 15) * 32;
  const float* lsrc = logits + (((long)(b * HEADS_ + h) * N_Q_) + q0) * N_O_;

  for (int i = 0; i < 64; ++i) {
    int f = i * 256 + t;
    L[f] = lsrc[f];
  }
  __syncthreads();

  // row softmax: row r handled by 8 consecutive lanes of one wave
  const int r = t >> 3, c8 = t & 7;
  float mx = -3.0e38f;
  for (int i = 0; i < 64; ++i) mx = fmaxf(mx, L[r * 512 + c8 + 8 * i]);
#pragma unroll
  for (int m = 1; m < 8; m <<= 1) mx = fmaxf(mx, __shfl_xor(mx, m, 8));
  float sm = 0.f;
  for (int i = 0; i < 64; ++i) {
    int idx = r * 512 + c8 + 8 * i;
    float e = expf(L[idx] - mx);
    L[idx] = e;
    sm += e;
  }
#pragma unroll
  for (int m = 1; m < 8; m <<= 1) sm += __shfl_xor(sm, m, 8);
  float inv = 1.f / sm;
  for (int i = 0; i < 64; ++i) L[r * 512 + c8 + 8 * i] *= inv;
  __syncthreads();

  // WMMA jobs
  const int w = t >> 5, l = t & 31, lm = l & 15, lg = l >> 4;
  const int which = w >> 2, mt = (w >> 1) & 1, nt = w & 1;
  v8f c = vzero8();
  const float* vb = v + (long)b * N_O_ * HD_ + h * HDIM_ + nt * 16 + lm;
  const int mrow = mt * 16 + lm;
  for (int kb = 0; kb < 128; ++kb) {
    int k0 = kb * 4 + 2 * lg;
    v2f a = {L[mrow * 512 + k0], L[mrow * 512 + k0 + 1]};
    float b0 = vb[(long)k0 * HD_];
    float b1 = vb[(long)(k0 + 1) * HD_];
    if (which) { b0 *= b0; b1 *= b1; }
    v2f bb = {b0, b1};
    c = wmma_f32(a, bb, c);
  }
  float* dst = which ? ev2 : hq;
#pragma unroll
  for (int j = 0; j < 8; ++j) {
    int q = q0 + mt * 16 + j + lg * 8;
    int d = nt * 16 + lm;
    dst[((long)b * N_Q_ + q) * HD_ + h * HDIM_ + d] = c[j];
  }
}

// ---------------------------------------------------------------------------
// K4: mean = hq @ ow + ob ; var_out = softplus((ev2 - hq^2) @ vw + vb)
// grid: (B*N_Q/16) row-tiles x 2 matrices; 8 waves = 8 N-tiles of 128 cols
// ---------------------------------------------------------------------------
__global__ __launch_bounds__(256) void k_out(
    const float* __restrict__ hq, const float* __restrict__ ev2,
    const float* __restrict__ ow, const float* __restrict__ ob,
    const float* __restrict__ vw, const float* __restrict__ vbias,
    float* __restrict__ out) {
  __shared__ float A[16 * 256];
  const int t = threadIdx.x;
  const int which = blockIdx.x & 1;
  const long row0 = (long)(blockIdx.x >> 1) * 16;

#pragma unroll
  for (int i = 0; i < 16; ++i) {
    int f = i * 256 + t;
    float x = hq[row0 * 256 + f];
    A[f] = which ? (ev2[row0 * 256 + f] - x * x) : x;
  }
  __syncthreads();

  const float* W = which ? vw : ow;
  const float* Bb = which ? vbias : ob;
  const int w = t >> 5, l = t & 31, lm = l & 15, lg = l >> 4;
  const int n = w * 16 + lm;

  v8f c = vzero8();
  for (int kb = 0; kb < 64; ++kb) {
    int k0 = kb * 4 + 2 * lg;
    v2f a = {A[lm * 256 + k0], A[lm * 256 + k0 + 1]};
    v2f bb = {W[k0 * ODIM_ + n], W[(k0 + 1) * ODIM_ + n]};
    c = wmma_f32(a, bb, c);
  }
  float bias = Bb[n];
  float* dst = out + (long)which * (B_ * N_Q_ * ODIM_);
#pragma unroll
  for (int j = 0; j < 8; ++j) {
    int m = j + lg * 8;
    float x = c[j] + bias;
    if (which) x = fmaxf(x, 0.f) + log1pf(expf(-fabsf(x)));  // softplus
    dst[(row0 + m) * ODIM_ + n] = x;
  }
}

// ---------------------------------------------------------------------------
extern "C" void kernel_launch(void* const* d_in, const int* in_sizes, int n_in,
                              void* d_out, int out_size, void* d_ws,
                              size_t ws_size, hipStream_t stream) {
  (void)in_sizes; (void)n_in; (void)out_size; (void)ws_size;
  const float* h_obs     = (const float*)d_in[0];
  const float* pos_obs   = (const float*)d_in[1];
  const float* pos_query = (const float*)d_in[2];
  const float* fw1       = (const float*)d_in[3];
  const float* fb1       = (const float*)d_in[4];
  const float* fw2       = (const float*)d_in[5];
  const float* fb2       = (const float*)d_in[6];
  const float* log_sigma = (const float*)d_in[7];
  const float* kw1       = (const float*)d_in[8];
  const float* kb1       = (const float*)d_in[9];
  const float* kw2       = (const float*)d_in[10];
  const float* kb2       = (const float*)d_in[11];
  const float* ow        = (const float*)d_in[12];
  const float* ob        = (const float*)d_in[13];
  const float* vw        = (const float*)d_in[14];
  const float* vb        = (const float*)d_in[15];

  float* ws    = (float*)d_ws;
  float* vbuf  = ws;                                      // B*N_O*HD
  float* kw1p  = vbuf + (long)B_ * N_O_ * HD_;            // 12*256
  float* kw2p  = kw1p + 12 * 256;                         // 256*16
  float* lgts  = kw2p + 256 * 16;                         // B*H*Nq*No
  float* hq    = lgts + (long)B_ * HEADS_ * N_Q_ * N_O_;  // B*Nq*HD
  float* ev2   = hq + (long)B_ * N_Q_ * HD_;              // B*Nq*HD

  k_init<<<1, 256, 0, stream>>>(kw1, kw2, kw1p, kw2p);
  k_feature_mlp<<<B_ * N_O_ / 16, 256, 0, stream>>>(h_obs, fw1, fb1, fw2, fb2,
                                                    vbuf);
  k_pair_mlp<<<B_ * 32 * 8, 256, 0, stream>>>(pos_obs, pos_query, log_sigma,
                                              kb1, kb2, kw1p, kw2p, lgts);
  k_attn<<<B_ * HEADS_ * (N_Q_ / 32), 256, 32 * 512 * sizeof(float), stream>>>(
      lgts, vbuf, hq, ev2);
  k_out<<<(B_ * N_Q_ / 16) * 2, 256, 0, stream>>>(hq, ev2, ow, ob, vw, vb,
                                                  (float*)d_out);
}